// QuantizedSharedEmbedding_43353399885901
// MI455X (gfx1250) — compile-verified
//
#include <hip/hip_runtime.h>

// Problem constants (from the reference)
#define KDIM 2048      // embedding width
#define NG   16        // groups per row (K / 128)
#define BLOCK 256      // threads per block (8 wave32s)
#define PER_THREAD 8   // KDIM / BLOCK elements per thread (all within one group)

typedef int   int4v   __attribute__((ext_vector_type(4)));
typedef float float4v __attribute__((ext_vector_type(4)));

#if __has_builtin(__builtin_amdgcn_global_load_async_to_lds_b128)
#define HAVE_ASYNC_LDS 1
#else
#define HAVE_ASYNC_LDS 0
#endif

#if __has_builtin(__builtin_amdgcn_s_wait_asynccnt)
#define WAIT_ASYNC(n) __builtin_amdgcn_s_wait_asynccnt(n)
#elif HAVE_ASYNC_LDS
#define WAIT_ASYNC(n) asm volatile("s_wait_asynccnt %0" ::"i"(n) : "memory")
#else
#define WAIT_ASYNC(n) ((void)0)
#endif

// Async-copy this thread's 32-byte slice of a row: global -> LDS.
// The instruction's IOFFSET is added to BOTH the global and LDS addresses,
// so two issues at offsets 0 and 16 reuse one address pair.
__device__ __forceinline__ void issue_async_32B(const int* gsrc, int* ldst) {
#if HAVE_ASYNC_LDS
  // Builtin signature: (int4 addrspace(1)*, int4 addrspace(3)*, imm offset, imm cpol)
  typedef __attribute__((address_space(1))) int4v g_int4;
  typedef __attribute__((address_space(3))) int4v l_int4;
  g_int4* gp = (g_int4*)gsrc;
  l_int4* lp = (l_int4*)ldst;
  __builtin_amdgcn_global_load_async_to_lds_b128(gp, lp, 0, 0);
  __builtin_amdgcn_global_load_async_to_lds_b128(gp, lp, 16, 0);
#else
  // Fallback: synchronous vectorized copy through VGPRs (still b128 traffic).
  ((int4v*)ldst)[0] = ((const int4v*)gsrc)[0];
  ((int4v*)ldst)[1] = ((const int4v*)gsrc)[1];
#endif
}

__global__ __launch_bounds__(BLOCK) void qembed_dequant_kernel(
    const int* __restrict__ x,       // [n_tokens]
    const int* __restrict__ qw,      // [VOCAB, KDIM] codes in [0,16)
    const float* __restrict__ sc,    // [VOCAB, NG]
    const int* __restrict__ zp,      // [VOCAB, NG]
    float* __restrict__ out,         // [n_tokens, KDIM]
    int n_tokens) {
  // Double-buffered row staging: 2 * 8 KB = 16 KB of the 320 KB WGP LDS.
  __shared__ __align__(16) int buf[2][KDIM];

  const int t = threadIdx.x;            // 0..255
  const int g = t >> 4;                 // (t*8)/128: this thread's quant group
  const int elem = t * PER_THREAD;      // base element within the row
  const unsigned stride = gridDim.x;

  int token = (int)blockIdx.x;
  if (token >= n_tokens) return;        // uniform per block

  // Prologue: start DMA of the first token's row into buffer 0.
  {
    const long long row0 = x[token];
    issue_async_32B(qw + row0 * KDIM + elem, &buf[0][elem]);
  }

  int cur = 0;
  for (; token < n_tokens; token += stride) {
    const long long row = x[token];

    // Prefetch the next token's row into the other buffer, then wait so that
    // only the 2 prefetch issues remain outstanding => current buffer ready.
    const int next = token + (int)stride;
    if (next < n_tokens) {
      const long long nrow = x[next];
      issue_async_32B(qw + nrow * KDIM + elem, &buf[cur ^ 1][elem]);
      WAIT_ASYNC(2);
    } else {
      WAIT_ASYNC(0);
    }

    // Per-(row,group) dequant parameters; fold zero*scale so each element
    // is a single v_fma after int->float conversion.
    const float s = sc[row * NG + g];
    const float nzs = -(float)zp[row * NG + g] * s;

    const int4v a = *(const int4v*)&buf[cur][elem];
    const int4v b = *(const int4v*)&buf[cur][elem + 4];

    float4v o0, o1;
    o0.x = fmaf((float)a.x, s, nzs);
    o0.y = fmaf((float)a.y, s, nzs);
    o0.z = fmaf((float)a.z, s, nzs);
    o0.w = fmaf((float)a.w, s, nzs);
    o1.x = fmaf((float)b.x, s, nzs);
    o1.y = fmaf((float)b.y, s, nzs);
    o1.z = fmaf((float)b.z, s, nzs);
    o1.w = fmaf((float)b.w, s, nzs);

    // Streaming (non-temporal) b128 stores: keep L2 for the gathered table.
    float* op = out + (long long)token * KDIM + elem;
    __builtin_nontemporal_store(o0, (float4v*)op);
    __builtin_nontemporal_store(o1, (float4v*)(op + 4));

    cur ^= 1;
  }
}

extern "C" void kernel_launch(void* const* d_in, const int* in_sizes, int n_in,
                              void* d_out, int out_size, void* d_ws, size_t ws_size,
                              hipStream_t stream) {
  const int*   x  = (const int*)d_in[0];
  const int*   qw = (const int*)d_in[1];
  const float* sc = (const float*)d_in[2];
  const int*   zp = (const int*)d_in[3];
  float* out = (float*)d_out;

  const int n_tokens = in_sizes[0];            // B*S = 32768
  int grid = n_tokens < 4096 ? n_tokens : 4096; // persistent blocks, ~8 tokens each
  if (grid < 1) grid = 1;

  hipLaunchKernelGGL(qembed_dequant_kernel, dim3(grid), dim3(BLOCK), 0, stream,
                     x, qw, sc, zp, out, n_tokens);
}